// LDS_LR_34265249087933
// MI455X (gfx1250) — compile-verified
//
#include <hip/hip_runtime.h>

// Problem constants (match reference)
#define BSZ_   4
#define SEQ_   8192
#define DIN_   256
#define DOUT_  256
#define STATE_ 512
#define LCH    32              // chunk length (time steps)
#define NCH    (SEQ_ / LCH)    // 256 chunks
#define XROWS  (LCH + 16)      // 16 pad rows for AR taps / tile alignment

// Workspace layout (float offsets)
#define OFF_WUT 0                      // WuT  [512][256]  (B1@B2)^T
#define OFF_C1T 131072                 // C1T  [64][512]   (rank padded)
#define OFF_C2T 163840                 // C2T  [256][64]
#define OFF_M1B 180224                 // M1B  [5][64][256]
#define OFF_M2B 262144                 // M2B  [5][256][64]
#define OFF_S   344064                 // chunk suffix sums [B][NC][512]
#define OFF_HIN 868352                 // chunk initial states [B][NC][512]

typedef float v2f __attribute__((ext_vector_type(2)));
typedef float v8f __attribute__((ext_vector_type(8)));

// One 16x16 output tile, K-loop of V_WMMA_F32_16X16X4_F32.
// A: row-major (M x K), pre-offset to (row0, 0), leading dim aLd.
// BT: B stored transposed (N x K) row-major, pre-offset to (col0, 0), leading dim bLd.
// Per ISA layout: lanes 0-15 / 16-31 both cover M(N)=0..15; lane half selects K pair.
__device__ __forceinline__ v8f wmma_row(const float* __restrict__ A, int aLd,
                                        const float* __restrict__ BT, int bLd,
                                        int K, v8f acc, int lane) {
  const int n15 = lane & 15;
  const int kh2 = (lane >> 4) << 1;           // 0 or 2
  const float* ap = A + n15 * aLd + kh2;
  const float* bp = BT + n15 * bLd + kh2;
  for (int k0 = 0; k0 < K; k0 += 4) {
    v2f a = *(const v2f*)(ap + k0);           // K = k0+kh2, k0+kh2+1  (ds/global b64)
    v2f b = *(const v2f*)(bp + k0);
    acc = __builtin_amdgcn_wmma_f32_16x16x4_f32(false, a, false, b,
                                                (short)0, acc, false, false);
  }
  return acc;
}

// C/D 16x16 f32 tile layout: VGPR v, lanes 0-15 -> M=v, lanes 16-31 -> M=8+v; N=lane&15.
__device__ __forceinline__ void store_tile(float* __restrict__ D, int dLd,
                                           v8f acc, int lane) {
  const int col = lane & 15;
  const int rb = (lane >> 4) * 8;
#pragma unroll
  for (int v = 0; v < 8; ++v) D[(rb + v) * dLd + col] = acc[v];
}

// ---------------------------------------------------------------- prep weights
__global__ __launch_bounds__(256) void p0_prep(const float* __restrict__ B1,
                                               const float* __restrict__ B2,
                                               const float* __restrict__ C1,
                                               const float* __restrict__ C2,
                                               const float* __restrict__ M1,
                                               const float* __restrict__ M2,
                                               float* __restrict__ ws) {
  int idx = blockIdx.x * 256 + threadIdx.x;   // grid covers 131072
  if (idx < 512 * 256) {                      // WuT[s][d] = sum_r B1[d][r]*B2[r][s]
    int s = idx >> 8, d = idx & 255;
    float acc = 0.f;
    for (int r = 0; r < 50; ++r) acc = fmaf(B1[d * 50 + r], B2[r * 512 + s], acc);
    ws[OFF_WUT + idx] = acc;
  }
  if (idx < 64 * 512) {                       // C1T[r][s] = C1[s][r], pad r>=50
    int r = idx >> 9, s = idx & 511;
    ws[OFF_C1T + idx] = (r < 50) ? C1[s * 50 + r] : 0.f;
  }
  if (idx < 256 * 64) {                       // C2T[o][r] = C2[r][o]
    int o = idx >> 6, r = idx & 63;
    ws[OFF_C2T + idx] = (r < 50) ? C2[r * 256 + o] : 0.f;
  }
  if (idx < 5 * 64 * 256) {                   // M1B[i][r][d] = M1[r][d][i]
    int i = idx >> 14, rem = idx & 16383;
    int r = rem >> 8, d = rem & 255;
    ws[OFF_M1B + idx] = (r < 50) ? M1[(r * 256 + d) * 5 + i] : 0.f;
  }
  if (idx < 5 * 256 * 64) {                   // M2B[i][o][r] = M2[o][r][i]
    int i = idx >> 14, rem = idx & 16383;
    int o = rem >> 6, r = rem & 63;
    ws[OFF_M2B + idx] = (r < 50) ? M2[(o * 50 + r) * 5 + i] : 0.f;
  }
}

// ------------------------------------------------- phase 1: chunk suffix sums
__global__ __launch_bounds__(256) void p1_suffix(const float* __restrict__ x,
                                                 const float* __restrict__ A,
                                                 float* __restrict__ ws) {
  __shared__ float XL[LCH * DIN_];            // 32x256
  __shared__ float UL[LCH * STATE_];          // 32x512
  const int c = blockIdx.x, b = blockIdx.y;
  const int t0 = c * LCH;
  const int tid = threadIdx.x;
  const int wave = tid >> 5, lane = tid & 31;
  const float* WuT = ws + OFF_WUT;

  const float* xb = x + ((size_t)b * SEQ_ + t0) * DIN_;
  for (int i4 = tid; i4 < LCH * (DIN_ / 4); i4 += 256)
    ((float4*)XL)[i4] = ((const float4*)xb)[i4];
  __syncthreads();

  // u = x @ Wu : 2x32 tiles of 16x16, 8 tiles/wave
  for (int tt = wave; tt < 64; tt += 8) {
    int tr = tt >> 5, tc = tt & 31;
    v8f acc = {};
    acc = wmma_row(XL + tr * 16 * DIN_, DIN_, WuT + tc * 16 * DIN_, DIN_,
                   DIN_, acc, lane);
    store_tile(UL + tr * 16 * STATE_ + tc * 16, STATE_, acc, lane);
  }
  __syncthreads();

  // local scan from 0 -> s_c = sum_k A^(L-1-k) u_k
  int s0 = tid, s1 = tid + 256;
  float a0 = A[s0], a1 = A[s1];
  float v0 = 0.f, v1 = 0.f;
  for (int k = 0; k < LCH; ++k) {
    v0 = fmaf(a0, v0, UL[k * STATE_ + s0]);
    v1 = fmaf(a1, v1, UL[k * STATE_ + s1]);
  }
  float* sdst = ws + OFF_S + ((size_t)(b * NCH + c)) * STATE_;
  sdst[s0] = v0;
  sdst[s1] = v1;
}

// --------------------------------------- phase 2: serial chunk-level carry scan
__global__ __launch_bounds__(512) void p2_carry(const float* __restrict__ A,
                                                const float* __restrict__ h0,
                                                float* __restrict__ ws) {
  const int s = threadIdx.x;                  // one thread per state
  float a = A[s];
  float aL = 1.f;
  for (int k = 0; k < LCH; ++k) aL *= a;      // A^L
  const float* sws = ws + OFF_S;
  float* hin = ws + OFF_HIN;
  for (int b = 0; b < BSZ_; ++b) {
    float h = h0[s];
    for (int c = 0; c < NCH; ++c) {
      size_t o = ((size_t)(b * NCH + c)) * STATE_ + s;
      hin[o] = h;                             // state entering chunk c
      h = fmaf(aL, h, sws[o]);
    }
  }
}

// -------------------------------- phase 3: u -> scan -> C1/C2 + AR -> output
__global__ __launch_bounds__(256) void p3_main(const float* __restrict__ x,
                                               const float* __restrict__ A,
                                               const float* __restrict__ ws,
                                               float* __restrict__ out) {
  __shared__ float XL[XROWS * DIN_];          // 48x256 (16 pad rows of history)
  __shared__ float UL[LCH * STATE_];          // u, then h, then pext (5*48*64<=16384)
  __shared__ float GL[LCH * 64];              // g = h@C1 (rank padded to 64)
  const int c = blockIdx.x, b = blockIdx.y;
  const int t0 = c * LCH;
  const int tid = threadIdx.x;
  const int wave = tid >> 5, lane = tid & 31;
  const float* WuT = ws + OFF_WUT;
  const float* C1T = ws + OFF_C1T;
  const float* C2T = ws + OFF_C2T;
  const float* M1B = ws + OFF_M1B;
  const float* M2B = ws + OFF_M2B;

  // load x rows [t0-16, t0+32), zero-pad t<0
  const float* xb = x + (size_t)b * SEQ_ * DIN_;
  for (int i4 = tid; i4 < XROWS * (DIN_ / 4); i4 += 256) {
    int r = i4 >> 6, q = i4 & 63;
    int t = t0 - 16 + r;
    float4 val = make_float4(0.f, 0.f, 0.f, 0.f);
    if (t >= 0) val = ((const float4*)(xb + (size_t)t * DIN_))[q];
    ((float4*)(XL + r * DIN_))[q] = val;
  }
  __syncthreads();

  // u = x @ Wu
  for (int tt = wave; tt < 64; tt += 8) {
    int tr = tt >> 5, tc = tt & 31;
    v8f acc = {};
    acc = wmma_row(XL + (16 + tr * 16) * DIN_, DIN_, WuT + tc * 16 * DIN_, DIN_,
                   DIN_, acc, lane);
    store_tile(UL + tr * 16 * STATE_ + tc * 16, STATE_, acc, lane);
  }
  __syncthreads();

  // diagonal scan: h_t = A*h_{t-1} + u_t, h overwrites u in LDS
  {
    const float* hin = ws + OFF_HIN + ((size_t)(b * NCH + c)) * STATE_;
    int s0 = tid, s1 = tid + 256;
    float a0 = A[s0], a1 = A[s1];
    float hA = hin[s0], hB = hin[s1];
    for (int k = 0; k < LCH; ++k) {
      float* row = UL + k * STATE_;
      hA = fmaf(a0, hA, row[s0]); row[s0] = hA;
      hB = fmaf(a1, hB, row[s1]); row[s1] = hB;
    }
  }
  __syncthreads();

  // g = h @ C1 : 2x4 tiles, one per wave
  {
    int tr = wave >> 2, tc = wave & 3;
    v8f acc = {};
    acc = wmma_row(UL + tr * 16 * STATE_, STATE_, C1T + tc * 16 * STATE_, STATE_,
                   STATE_, acc, lane);
    store_tile(GL + tr * 16 * 64 + tc * 16, 64, acc, lane);
  }
  __syncthreads();

  // pext_i = x_ext @ M1_i^T : 5 taps x 3x4 tiles, reuse UL
  for (int tt = wave; tt < 60; tt += 8) {
    int i = tt / 12, rem = tt % 12;
    int pr = rem >> 2, pc = rem & 3;
    v8f acc = {};
    acc = wmma_row(XL + pr * 16 * DIN_, DIN_, M1B + (i * 64 + pc * 16) * DIN_,
                   DIN_, DIN_, acc, lane);
    store_tile(UL + i * (XROWS * 64) + pr * 16 * 64 + pc * 16, 64, acc, lane);
  }
  __syncthreads();

  // out = g @ C2 + sum_i shift_i(pext_i) @ M2_i^T : 2x16 tiles
  float* ob = out + ((size_t)b * SEQ_ + t0) * DOUT_;
  for (int tt = wave; tt < 32; tt += 8) {
    int tr = tt >> 4, oc = tt & 15;
    v8f acc = {};
    acc = wmma_row(GL + tr * 16 * 64, 64, C2T + oc * 16 * 64, 64, 64, acc, lane);
    for (int i = 0; i < 5; ++i) {
      acc = wmma_row(UL + i * (XROWS * 64) + (16 - i + tr * 16) * 64, 64,
                     M2B + (i * 256 + oc * 16) * 64, 64, 64, acc, lane);
    }
    const int col = lane & 15;
    const int rb = (lane >> 4) * 8;
#pragma unroll
    for (int v = 0; v < 8; ++v)
      ob[(size_t)(tr * 16 + rb + v) * DOUT_ + oc * 16 + col] = acc[v];
  }
}

extern "C" void kernel_launch(void* const* d_in, const int* in_sizes, int n_in,
                              void* d_out, int out_size, void* d_ws, size_t ws_size,
                              hipStream_t stream) {
  (void)in_sizes; (void)n_in; (void)out_size; (void)ws_size;
  const float* x  = (const float*)d_in[0];
  const float* h0 = (const float*)d_in[1];
  const float* A  = (const float*)d_in[2];
  const float* B1 = (const float*)d_in[3];
  const float* B2 = (const float*)d_in[4];
  const float* C1 = (const float*)d_in[5];
  const float* C2 = (const float*)d_in[6];
  const float* M1 = (const float*)d_in[7];
  const float* M2 = (const float*)d_in[8];
  float* ws = (float*)d_ws;
  float* out = (float*)d_out;

  p0_prep<<<512, 256, 0, stream>>>(B1, B2, C1, C2, M1, M2, ws);
  p1_suffix<<<dim3(NCH, BSZ_), 256, 0, stream>>>(x, A, ws);
  p2_carry<<<1, 512, 0, stream>>>(A, h0, ws);
  p3_main<<<dim3(NCH, BSZ_), 256, 0, stream>>>(x, A, ws, out);
}